// DWTPooling_33887291965641
// MI455X (gfx1250) — compile-verified
//
#include <hip/hip_runtime.h>
#include <hip/hip_bf16.h>

// DWT (Haar) 2x2 pooling, NHWC fp32.
//   in : (8, 512, 512, 64)   -> 512 MiB
//   out: (8, 256, 256, 256)  -> 512 MiB   (per channel c: out[...,4c+k] = {ll,lh,hl,hh})
//
// Pure streaming kernel: HBM-bandwidth bound (~1 GiB / 23.3 TB/s ~ 46 us floor).
// One thread per (output pixel, 4-channel group): 4 NT b128 loads, 4 NT b128
// stores, 48 dual-issued fp ops. 128 B of traffic per thread keeps the
// instruction-issue requirement ~4x below the bandwidth-floor demand.

#define B_  8u
#define H_  512u
#define W_  512u
#define C_  64u

// native clang vector type so nontemporal builtins accept it (lowers to b128)
typedef float v4f __attribute__((ext_vector_type(4)));

__global__ __launch_bounds__(256) void dwt_haar_pool_kernel(
    const float* __restrict__ x, float* __restrict__ out)
{
    // tid = pixel*16 + channel-group ; pixel = b*256*256 + h*256 + w (pow2 -> shifts)
    const unsigned tid = blockIdx.x * 256u + threadIdx.x;
    const unsigned t   = tid & 15u;              // channel group: channels 4t..4t+3
    const unsigned pix = tid >> 4;               // output pixel index
    const unsigned w   = pix & 255u;             // 0..255
    const unsigned h   = (pix >> 8) & 255u;      // 0..255
    const unsigned b   = pix >> 16;              // 0..7

    // input strides in floats
    const size_t rowStride = (size_t)W_ * C_;          // 32768 floats (128 KiB)
    const size_t imgStride = (size_t)H_ * rowStride;   // 16777216 floats

    const size_t base = (size_t)b * imgStride
                      + (size_t)(2u * h) * rowStride
                      + (size_t)(2u * w) * C_
                      + 4u * t;                        // 16B aligned

    // a = x[b,2h,2w,4t..4t+3], b at +256B, c at +128KiB, d at +128KiB+256B
    // (all immediate offsets from one address -> one s_clause of 4 b128 NT loads)
    const v4f va = __builtin_nontemporal_load((const v4f*)(x + base));
    const v4f vb = __builtin_nontemporal_load((const v4f*)(x + base + C_));
    const v4f vc = __builtin_nontemporal_load((const v4f*)(x + base + rowStride));
    const v4f vd = __builtin_nontemporal_load((const v4f*)(x + base + rowStride + C_));

    // out base for this thread: pix*256 + 4*(4t) floats; channel 4t+j's
    // {ll,lh,hl,hh} quad sits at +16j bytes (immediate offsets 0/16/32/48).
    float* const ob = out + ((size_t)pix << 8) + ((size_t)t << 4);

#pragma unroll
    for (unsigned j = 0; j < 4; ++j) {
        const float s0 = va[j] + vb[j];
        const float d0 = va[j] - vb[j];
        const float s1 = vc[j] + vd[j];
        const float d1 = vc[j] - vd[j];
        v4f r;
        r.x = 0.5f * (s0 + s1);   // ll
        r.y = 0.5f * (s0 - s1);   // lh
        r.z = 0.5f * (d0 + d1);   // hl
        r.w = 0.5f * (d0 - d1);   // hh
        __builtin_nontemporal_store(r, (v4f*)(ob + 4u * j));
    }
}

extern "C" void kernel_launch(void* const* d_in, const int* in_sizes, int n_in,
                              void* d_out, int out_size, void* d_ws, size_t ws_size,
                              hipStream_t stream)
{
    const float* x  = (const float*)d_in[0];
    float* out      = (float*)d_out;

    // one thread per (output pixel, 4-channel group) = 8*256*256*16 = 8,388,608
    const unsigned nThreads = B_ * (H_ / 2u) * (W_ / 2u) * (C_ / 4u);
    const unsigned nBlocks  = nThreads / 256u;   // 32768

    dwt_haar_pool_kernel<<<nBlocks, 256, 0, stream>>>(x, out);
}